// FourierNN_81080392614130
// MI455X (gfx1250) — compile-verified
//
#include <hip/hip_runtime.h>
#include <math.h>

typedef __attribute__((ext_vector_type(2))) float v2f;
typedef __attribute__((ext_vector_type(4))) float v4f;
typedef __attribute__((ext_vector_type(8))) float v8f;

#define PI2 6.28318530717958647692f

// ---------------------------------------------------------------------------
// K0a: partial rfft along y: Cy[f][x][j] = sum_y B[x,y] e^{-2pi i j y/128},
//      j=0..15, f in {Bp, -Bt, Br}.  6144 threads.
// ---------------------------------------------------------------------------
__global__ void fnn_k0a_rffty(const float* __restrict__ Br, const float* __restrict__ Bp,
                              const float* __restrict__ Bt, float* __restrict__ Cy) {
    int tid = blockIdx.x * blockDim.x + threadIdx.x;       // 0..6143
    int f = tid >> 11;
    int x = (tid >> 4) & 127;
    int j = tid & 15;
    const float* B = (f == 0) ? Bp : (f == 1) ? Bt : Br;
    float sgn = (f == 1) ? -1.f : 1.f;
    const float* row = B + x * 128;
    float ar = 0.f, ai = 0.f;
    float w = PI2 * (float)j / 128.f;
    for (int y = 0; y < 128; ++y) {
        float s, c; __sincosf(w * (float)y, &s, &c);
        float v = sgn * row[y];
        ar += v * c;
        ai -= v * s;
    }
    Cy[tid * 2 + 0] = ar;
    Cy[tid * 2 + 1] = ai;
}

// ---------------------------------------------------------------------------
// K0b: fft along x for the 32 corner kx modes, assemble xr[3072].
//      1536 threads.
// ---------------------------------------------------------------------------
__global__ void fnn_k0b_fftx(const float* __restrict__ Cy, float* __restrict__ xr) {
    int tid = blockIdx.x * blockDim.x + threadIdx.x;       // 0..1535
    int f   = tid >> 9;
    int ki  = (tid >> 4) & 31;
    int j   = tid & 15;
    int kxl = ki & 15, hb = ki >> 4;
    float kxv = hb ? (float)(kxl - 16) : (float)kxl;       // fftfreq value
    float w = PI2 * kxv / 128.f;
    float dr = 0.f, di = 0.f;
    const float* base = Cy + (size_t)f * 128 * 16 * 2;
    for (int x = 0; x < 128; ++x) {
        float s, c; __sincosf(w * (float)x, &s, &c);
        float cr = base[(x * 16 + j) * 2 + 0];
        float ci = base[(x * 16 + j) * 2 + 1];
        dr += cr * c + ci * s;        // (cr + i ci) * e^{-i w x}
        di += ci * c - cr * s;
    }
    int idx = ((f * 2 + hb) * 16 + kxl) * 16 + j;
    xr[idx]        = dr;
    xr[1536 + idx] = di;
}

// ---------------------------------------------------------------------------
// K1: h = sigmoid(W1 @ xr + b1).  One wave per row, 1024 rows.
// ---------------------------------------------------------------------------
__global__ void fnn_k1_dense1(const float* __restrict__ W1, const float* __restrict__ b1,
                              const float* __restrict__ xr, float* __restrict__ h) {
    __shared__ float sx[3072];
    for (int i = threadIdx.x; i < 3072; i += 256) sx[i] = xr[i];
    __syncthreads();
    int wave = threadIdx.x >> 5, lane = threadIdx.x & 31;
    int row = blockIdx.x * 8 + wave;                       // 0..1023
    const float* wr = W1 + (size_t)row * 3072;
    float acc = 0.f;
    for (int t = 0; t < 24; ++t) {
        int o = lane * 4 + t * 128;
        v4f a = *(const v4f*)(wr + o);
        v4f b = *(const v4f*)(sx + o);
        acc += a.x * b.x + a.y * b.y + a.z * b.z + a.w * b.w;
    }
    for (int m = 16; m; m >>= 1) acc += __shfl_xor(acc, m, 32);
    if (lane == 0) {
        float z = acc + b1[row];
        h[row] = 1.f / (1.f + __expf(-z));
    }
}

// ---------------------------------------------------------------------------
// K2: complex output layer (bandwidth bound: 402 MB of W2).  Wave per row.
// ---------------------------------------------------------------------------
__global__ void fnn_k2_dense2(const float* __restrict__ W2r, const float* __restrict__ W2i,
                              const float* __restrict__ b2r, const float* __restrict__ b2i,
                              const float* __restrict__ h,
                              float* __restrict__ outr, float* __restrict__ outi) {
    __shared__ float sh[1024];
    for (int i = threadIdx.x; i < 1024; i += 256) sh[i] = h[i];
    __syncthreads();
    int wave = threadIdx.x >> 5, lane = threadIdx.x & 31;
    int row = blockIdx.x * 8 + wave;                       // 0..49151
    const float* wr = W2r + (size_t)row * 1024;
    const float* wi = W2i + (size_t)row * 1024;
    float ar = 0.f, ai = 0.f;
    for (int t = 0; t < 8; ++t) {
        int o = lane * 4 + t * 128;
        v4f hh = *(const v4f*)(sh + o);
        v4f a  = *(const v4f*)(wr + o);
        v4f b  = *(const v4f*)(wi + o);
        ar += a.x * hh.x + a.y * hh.y + a.z * hh.z + a.w * hh.w;
        ai += b.x * hh.x + b.y * hh.y + b.z * hh.z + b.w * hh.w;
    }
    for (int m = 16; m; m >>= 1) { ar += __shfl_xor(ar, m, 32); ai += __shfl_xor(ai, m, 32); }
    if (lane == 0) {
        outr[row] = ar + b2r[row];
        outi[row] = ai + b2i[row];
    }
}

// ---------------------------------------------------------------------------
// KA: contract kx (32 modes -> 128 x), produce plain and d/dx spectra,
//     written directly in the K=64 "complex-as-real" B-matrix layout that
//     stage B's WMMA consumes:  astb[arr][x][n(32)][k(64)]
//     n<16: Re out (kz=n), n>=16: Im out;  k<32: Ar/Ai, k>=32: -Ai/Ar.
//     arr 0..2 = plain(vec), arr 3..5 = d/dx(vec).  1/(128*128) folded here.
// ---------------------------------------------------------------------------
__global__ void fnn_ka_xcontract(const float* __restrict__ outr, const float* __restrict__ outi,
                                 float* __restrict__ astb) {
    int tid = blockIdx.x * 256 + threadIdx.x;              // 0..196607
    int vec = tid >> 16;
    int x   = (tid >> 9) & 127;
    int j   = (tid >> 4) & 31;                             // ky mode index
    int kz  = tid & 15;
    int crlo = (j >= 16) ? 1 : 0;
    float a0r = 0.f, a0i = 0.f, a1r = 0.f, a1i = 0.f;
    for (int ki = 0; ki < 32; ++ki) {
        int kxl = ki & 15, hb = ki >> 4;
        int cr = hb * 2 + crlo;
        int oidx = (vec * 4 + cr) * 4096 + kxl * 256 + (j & 15) * 16 + kz;
        float Fr = outr[oidx], Fi = outi[oidx];
        float kxv = hb ? (float)(kxl - 16) : (float)kxl;
        float th = PI2 * kxv * (float)x / 128.f;
        float s, c; __sincosf(th, &s, &c);
        a0r += c * Fr - s * Fi;                            // e^{+i th} * F
        a0i += c * Fi + s * Fr;
        a1r += kxv * (-c * Fi - s * Fr);                   // i*kx * e^{+i th} * F
        a1i += kxv * ( c * Fr - s * Fi);
    }
    const float scale = 1.f / 16384.f;                     // 1/(128*128)
    a0r *= scale; a0i *= scale; a1r *= scale; a1i *= scale;
    size_t b0 = (((size_t)vec * 128 + x) * 32) * 64;
    size_t b1 = (((size_t)(3 + vec) * 128 + x) * 32) * 64;
    int nR = kz, nI = 16 + kz, kR = j, kI = 32 + j;
    astb[b0 + (size_t)nR * 64 + kR] =  a0r;
    astb[b0 + (size_t)nI * 64 + kR] =  a0i;
    astb[b0 + (size_t)nR * 64 + kI] = -a0i;
    astb[b0 + (size_t)nI * 64 + kI] =  a0r;
    astb[b1 + (size_t)nR * 64 + kR] =  a1r;
    astb[b1 + (size_t)nI * 64 + kR] =  a1i;
    astb[b1 + (size_t)nR * 64 + kI] = -a1i;
    astb[b1 + (size_t)nI * 64 + kI] =  a1r;
}

// ---------------------------------------------------------------------------
// KB: contract ky with WMMA f32 16x16x4.  D[y,n] = Ey[y,k(64)] x Bmat[k,n(32)]
//     per (arr, x) unit; arr 0..5 use plain e^{i th}, arr 6..8 use i*ky basis
//     on the plain spectra (d/dy).  Output G[arr9][x][y][ri2][kz16].
// ---------------------------------------------------------------------------
__global__ void fnn_kb_ycontract(const float* __restrict__ astb, float* __restrict__ G) {
    __shared__ float ey[128 * 64];                         // [y][k]
    int unit0 = blockIdx.x * 8;
    int arr = unit0 >> 7;                                  // uniform per block
    bool dy = (arr >= 6);
    for (int i = threadIdx.x; i < 128 * 64; i += 256) {
        int y = i >> 6, k = i & 63;
        int j = k & 31; bool im = (k >= 32);
        float jv = (j < 16) ? (float)j : (float)(j - 32);  // ky wavenumber
        float th = PI2 * jv * (float)y / 128.f;
        float s, c; __sincosf(th, &s, &c);
        ey[i] = dy ? (im ? jv * c : -jv * s)               // i*ky*e^{i th}
                   : (im ? s : c);                          // e^{i th}
    }
    __syncthreads();
    int wave = threadIdx.x >> 5, lane = threadIdx.x & 31;
    int hi = lane >> 4, ln = lane & 15;
    int unit = unit0 + wave;
    int x = unit & 127;
    int srcArr = (arr < 6) ? arr : (arr - 6);
    const float* bsrc = astb + (((size_t)srcArr * 128 + x) * 32) * 64;
    v2f bf[2][16];
    #pragma unroll
    for (int nt = 0; nt < 2; ++nt)
        #pragma unroll
        for (int t = 0; t < 16; ++t)
            bf[nt][t] = *(const v2f*)(bsrc + (size_t)(nt * 16 + ln) * 64 + 4 * t + 2 * hi);
    float* gbase = G + (((size_t)arr * 128 + x) * 128) * 32;
    for (int yt = 0; yt < 8; ++yt) {
        #pragma unroll
        for (int nt = 0; nt < 2; ++nt) {
            v8f acc = {0.f, 0.f, 0.f, 0.f, 0.f, 0.f, 0.f, 0.f};
            #pragma unroll
            for (int t = 0; t < 16; ++t) {
                v2f af = *(const v2f*)&ey[(yt * 16 + ln) * 64 + 4 * t + 2 * hi];
                acc = __builtin_amdgcn_wmma_f32_16x16x4_f32(
                        false, af, false, bf[nt][t], (short)0, acc, false, false);
            }
            int N = nt * 16 + ln;
            #pragma unroll
            for (int r = 0; r < 8; ++r) {
                int y = yt * 16 + r + 8 * hi;
                gbase[(size_t)y * 32 + N] = acc[r];
            }
        }
    }
}

// ---------------------------------------------------------------------------
// KC: contract kz (irfft z axis) with WMMA f32 16x16x4.
//     D[z(254), xy(16)] = BzT[z, k(32)] x G^T[k, xy]; 12 output channels.
//     1/254 and the half-spectrum weights folded into the basis; lane stores
//     contiguous z-runs so output writes stay vectorized.
// ---------------------------------------------------------------------------
__global__ void fnn_kc_zcontract(const float* __restrict__ G, float* __restrict__ out) {
    __shared__ float bz[256 * 32];                         // [z][k]
    int c = blockIdx.x >> 7;                               // channel 0..11
    bool dz = (c >= 3) && (((c - 3) % 3) == 2);
    for (int i = threadIdx.x; i < 256 * 32; i += 256) {
        int z = i >> 5, k = i & 31;
        int kz = k & 15; bool im = (k >= 16);
        float v = 0.f;
        if (z < 254) {
            float sW = (kz == 0) ? 1.f : 2.f;              // half-spectrum weight
            float th = PI2 * (float)kz * (float)z / 254.f;
            float s, ct; __sincosf(th, &s, &ct);
            if (!dz) v = (im ? -s : ct) * sW * (1.f / 254.f);
            else     v = (im ? -ct : -s) * sW * (float)kz * (1.f / 254.f);
        }
        bz[i] = v;
    }
    __syncthreads();
    int wave = threadIdx.x >> 5, lane = threadIdx.x & 31;
    int hi = lane >> 4, ln = lane & 15;
    int xyTile = (blockIdx.x & 127) * 8 + wave;            // 0..1023
    int ga;
    if (c < 3) ga = c;
    else { int v = (c - 3) / 3, d = (c - 3) % 3; ga = (d == 0) ? 3 + v : (d == 1) ? 6 + v : v; }
    const float* grow = G + ((size_t)ga * 16384 + (size_t)xyTile * 16 + ln) * 32;
    v2f bfr[8];
    #pragma unroll
    for (int t = 0; t < 8; ++t) bfr[t] = *(const v2f*)(grow + 4 * t + 2 * hi);
    float* outc = out + (size_t)c * 16384 * 254;
    int xy = xyTile * 16 + ln;
    for (int zt = 0; zt < 16; ++zt) {
        v8f acc = {0.f, 0.f, 0.f, 0.f, 0.f, 0.f, 0.f, 0.f};
        #pragma unroll
        for (int t = 0; t < 8; ++t) {
            v2f af = *(const v2f*)&bz[(zt * 16 + ln) * 32 + 4 * t + 2 * hi];
            acc = __builtin_amdgcn_wmma_f32_16x16x4_f32(
                    false, af, false, bfr[t], (short)0, acc, false, false);
        }
        int zb = zt * 16 + 8 * hi;
        float* p = outc + (size_t)xy * 254 + zb;
        if (zt < 15) {
            *(v2f*)(p + 0) = (v2f){acc[0], acc[1]};
            *(v2f*)(p + 2) = (v2f){acc[2], acc[3]};
            *(v2f*)(p + 4) = (v2f){acc[4], acc[5]};
            *(v2f*)(p + 6) = (v2f){acc[6], acc[7]};
        } else {
            #pragma unroll
            for (int r = 0; r < 8; ++r) if (zb + r < 254) p[r] = acc[r];
        }
    }
}

// ---------------------------------------------------------------------------
extern "C" void kernel_launch(void* const* d_in, const int* in_sizes, int n_in,
                              void* d_out, int out_size, void* d_ws, size_t ws_size,
                              hipStream_t stream) {
    (void)in_sizes; (void)n_in; (void)out_size; (void)ws_size;
    const float* Br  = (const float*)d_in[0];
    const float* Bp  = (const float*)d_in[1];
    const float* Bt  = (const float*)d_in[2];
    const float* W1  = (const float*)d_in[3];
    const float* b1  = (const float*)d_in[4];
    const float* W2r = (const float*)d_in[5];
    const float* W2i = (const float*)d_in[6];
    const float* b2r = (const float*)d_in[7];
    const float* b2i = (const float*)d_in[8];

    float* ws   = (float*)d_ws;
    float* xr   = ws;                 //    3072
    float* h    = ws + 3072;          //    1024
    float* Cy   = ws + 4096;          //   12288
    float* orr  = ws + 16384;         //   49152
    float* oii  = ws + 65536;         //   49152
    float* astb = ws + 114688;        // 1572864  (6 x 128 x 32 x 64)
    float* G    = ws + 1687552;       // 4718592  (9 x 128 x 128 x 32)

    fnn_k0a_rffty   <<<  24, 256, 0, stream>>>(Br, Bp, Bt, Cy);
    fnn_k0b_fftx    <<<   6, 256, 0, stream>>>(Cy, xr);
    fnn_k1_dense1   <<< 128, 256, 0, stream>>>(W1, b1, xr, h);
    fnn_k2_dense2   <<<6144, 256, 0, stream>>>(W2r, W2i, b2r, b2i, h, orr, oii);
    fnn_ka_xcontract<<< 768, 256, 0, stream>>>(orr, oii, astb);
    fnn_kb_ycontract<<< 144, 256, 0, stream>>>(astb, G);
    fnn_kc_zcontract<<<1536, 256, 0, stream>>>(G, (float*)d_out);
}